// multiHeadGAT_90220083019814
// MI455X (gfx1250) — compile-verified
//
#include <hip/hip_runtime.h>

#define NN 50000
#define NE 800000
#define NF 128

typedef float v2f __attribute__((ext_vector_type(2)));
typedef float v8f __attribute__((ext_vector_type(8)));

// ---- ordered-uint encoding for float atomic max ----
__device__ __forceinline__ unsigned f2o(float f) {
    unsigned u = __float_as_uint(f);
    return (u & 0x80000000u) ? ~u : (u | 0x80000000u);
}
__device__ __forceinline__ float o2f(unsigned e) {
    unsigned u = (e & 0x80000000u) ? (e ^ 0x80000000u) : ~e;
    return __uint_as_float(u);
}

// ---- zero den + gmax (first 200064 dwords of ws) ----
__global__ void __launch_bounds__(256) init_kernel(unsigned* __restrict__ ws) {
    int i = blockIdx.x * 256 + threadIdx.x;
    if (i < 200064) ws[i] = 0u;
}

// ---- H = V @ W_cat via V_WMMA_F32_16X16X4_F32; one wave = 16x128 tile ----
__global__ void __launch_bounds__(256) node_transform(const float* __restrict__ V,
                                                      const float* __restrict__ W,
                                                      float* __restrict__ H) {
    const int lane = threadIdx.x & 31;
    const int tile = blockIdx.x * 8 + (threadIdx.x >> 5);
    if (tile >= NN / 16) return;
    const int m0   = tile * 16;
    const int half = lane >> 4;   // lanes 16-31 hold K=2,3 (A) / K=2,3 rows (B)
    const int l15  = lane & 15;

    const float* Ap = V + (m0 + l15) * NF + half * 2;   // A: row m, K pair
    const float* Bp[8];
#pragma unroll
    for (int nt = 0; nt < 8; ++nt) {
        int ncol = nt * 16 + l15;                       // output column = h*32+kk
        // W_edge[h][k][kk]: base = h*4096 + kk, element k at +k*32
        Bp[nt] = W + ((ncol >> 5) << 12) + (ncol & 31) + half * 64;
    }
    v8f c[8] = {};
    for (int k0 = 0; k0 < NF; k0 += 4) {
        v2f a = *(const v2f*)(Ap + k0);
#pragma unroll
        for (int nt = 0; nt < 8; ++nt) {
            v2f b;
            b.x = Bp[nt][k0 * 32];
            b.y = Bp[nt][k0 * 32 + 32];
            c[nt] = __builtin_amdgcn_wmma_f32_16x16x4_f32(
                false, a, false, b, (short)0, c[nt], false, false);
        }
    }
    // C layout: lanes 0-15 rows 0-7, lanes 16-31 rows 8-15; col = nt*16 + l15
#pragma unroll
    for (int nt = 0; nt < 8; ++nt) {
        float* Op = H + (m0 + half * 8) * NF + nt * 16 + l15;
#pragma unroll
        for (int j = 0; j < 8; ++j) Op[j * NF] = c[nt][j];
    }
}

// ---- a_s[n,h] = sum_k H[n,h,k]*w_s[h,k], a_r likewise; one wave per node ----
__global__ void __launch_bounds__(256) node_aux(const float* __restrict__ H,
                                                const float* __restrict__ aw,
                                                float* __restrict__ a_s,
                                                float* __restrict__ a_r) {
    const int lane = threadIdx.x & 31;
    const int n = blockIdx.x * 8 + (threadIdx.x >> 5);
    if (n >= NN) return;
    float4 hv = ((const float4*)H)[n * 32 + lane];      // cols 4*lane .. 4*lane+3
    int h  = lane >> 3;                                 // head for this lane
    int k4 = lane & 7;                                  // k-quad within head
    float4 wsv = ((const float4*)(aw + h * 192))[k4];
    float4 wrv = ((const float4*)(aw + h * 192 + 32))[k4];
    float ps = fmaf(hv.x, wsv.x, fmaf(hv.y, wsv.y, fmaf(hv.z, wsv.z, hv.w * wsv.w)));
    float pr = fmaf(hv.x, wrv.x, fmaf(hv.y, wrv.y, fmaf(hv.z, wrv.z, hv.w * wrv.w)));
#pragma unroll
    for (int off = 4; off >= 1; off >>= 1) {            // reduce within 8-lane head group
        ps += __shfl_xor(ps, off, 32);
        pr += __shfl_xor(pr, off, 32);
    }
    if (k4 == 0) { a_s[n * 4 + h] = ps; a_r[n * 4 + h] = pr; }
}

// ---- edge pass: MODE 0 = scores + global max (optionally cache), MODE 1 = exp + den add ----
template <int MODE>
__global__ void __launch_bounds__(256) edge_kernel(const float* __restrict__ E,
                                                   const int* __restrict__ edges,
                                                   const float* __restrict__ aw,
                                                   const float* __restrict__ ab,
                                                   const float* __restrict__ a_s,
                                                   const float* __restrict__ a_r,
                                                   float* __restrict__ scores,
                                                   int use_cache,
                                                   unsigned* __restrict__ gmax,
                                                   float* __restrict__ den) {
    const int lane = threadIdx.x & 31;
    float4 we[4];
#pragma unroll
    for (int h = 0; h < 4; ++h)                          // w_e[h][4*lane..+3]
        we[h] = ((const float4*)aw)[h * 48 + 16 + lane];
    float bb = (lane < 4) ? ab[lane] : 0.f;
    float mh = 0.f;
    if (MODE == 1 && lane < 4) mh = o2f(gmax[lane]);
    float lmax = -3.402823466e38f;

    const int gw = blockIdx.x * 8 + (threadIdx.x >> 5);
    const int nw = gridDim.x * 8;
    for (int e = gw; e < NE; e += nw) {
        float4 ev = ((const float4*)E)[e * 32 + lane];   // coalesced 512B per wave
        float d0 = fmaf(ev.x, we[0].x, fmaf(ev.y, we[0].y, fmaf(ev.z, we[0].z, ev.w * we[0].w)));
        float d1 = fmaf(ev.x, we[1].x, fmaf(ev.y, we[1].y, fmaf(ev.z, we[1].z, ev.w * we[1].w)));
        float d2 = fmaf(ev.x, we[2].x, fmaf(ev.y, we[2].y, fmaf(ev.z, we[2].z, ev.w * we[2].w)));
        float d3 = fmaf(ev.x, we[3].x, fmaf(ev.y, we[3].y, fmaf(ev.z, we[3].z, ev.w * we[3].w)));
#pragma unroll
        for (int off = 16; off >= 1; off >>= 1) {
            d0 += __shfl_xor(d0, off, 32);
            d1 += __shfl_xor(d1, off, 32);
            d2 += __shfl_xor(d2, off, 32);
            d3 += __shfl_xor(d3, off, 32);
        }
        if (lane < 4) {
            float dv = lane == 0 ? d0 : lane == 1 ? d1 : lane == 2 ? d2 : d3;
            int s = edges[2 * e];
            int r = edges[2 * e + 1];
            float sc = dv + a_s[s * 4 + lane] + a_r[r * 4 + lane] + bb;
            sc = sc > 0.f ? sc : 0.2f * sc;              // leaky_relu 0.2
            if (MODE == 0) {
                if (use_cache) scores[lane * NE + e] = sc;
                lmax = fmaxf(lmax, sc);
            } else {
                atomicAdd(&den[s * 4 + lane], __expf(sc - mh));
            }
        }
    }
    if (MODE == 0) {
        __shared__ unsigned smax[4];
        if (threadIdx.x < 4) smax[threadIdx.x] = 0u;
        __syncthreads();
        if (lane < 4) atomicMax(&smax[lane], f2o(lmax));
        __syncthreads();
        if (threadIdx.x < 4) atomicMax(&gmax[threadIdx.x], smax[threadIdx.x]);
    }
}

// ---- pass2 fast path: read cached scores ----
__global__ void __launch_bounds__(256) pass2_cached(const float* __restrict__ scores,
                                                    const int* __restrict__ edges,
                                                    const unsigned* __restrict__ gmax,
                                                    float* __restrict__ den) {
    int e = blockIdx.x * 256 + threadIdx.x;
    int h = blockIdx.y;
    if (e >= NE) return;
    float att = __expf(scores[h * NE + e] - o2f(gmax[h]));
    atomicAdd(&den[edges[2 * e] * 4 + h], att);
}

// ---- out = H * den/(den+eps), in place ----
__global__ void __launch_bounds__(256) finalize(float* __restrict__ out,
                                                const float* __restrict__ den) {
    int i = blockIdx.x * 256 + threadIdx.x;
    if (i >= NN * NF) return;
    int n = i >> 7;
    int h = (i >> 5) & 3;
    float dn = den[n * 4 + h];
    out[i] *= dn / (dn + 1e-8f);
}

extern "C" void kernel_launch(void* const* d_in, const int* in_sizes, int n_in,
                              void* d_out, int out_size, void* d_ws, size_t ws_size,
                              hipStream_t stream) {
    const float* V     = (const float*)d_in[0];
    const float* E     = (const float*)d_in[1];
    const int*   edges = (const int*)d_in[2];
    const float* W     = (const float*)d_in[3];
    const float* aw    = (const float*)d_in[4];
    const float* ab    = (const float*)d_in[5];
    float* out = (float*)d_out;

    float* wsf      = (float*)d_ws;
    unsigned* gmax  = (unsigned*)d_ws;      // [0..3], padded to 64 dwords
    float* den      = wsf + 64;             // 200000 floats
    float* a_s      = wsf + 200064;         // 200000 floats
    float* a_r      = wsf + 400064;         // 200000 floats
    float* scores   = wsf + 600064;         // 3,200,000 floats (optional cache)
    int use_cache = (ws_size >= (size_t)(600064 + 4 * (size_t)NE) * 4) ? 1 : 0;

    init_kernel<<<(200064 + 255) / 256, 256, 0, stream>>>((unsigned*)d_ws);
    node_transform<<<(NN / 16 + 7) / 8, 256, 0, stream>>>(V, W, out);
    node_aux<<<(NN + 7) / 8, 256, 0, stream>>>(out, aw, a_s, a_r);
    edge_kernel<0><<<1024, 256, 0, stream>>>(E, edges, aw, ab, a_s, a_r,
                                             scores, use_cache, gmax, den);
    if (use_cache) {
        pass2_cached<<<dim3((NE + 255) / 256, 4), 256, 0, stream>>>(scores, edges, gmax, den);
    } else {
        edge_kernel<1><<<1024, 256, 0, stream>>>(E, edges, aw, ab, a_s, a_r,
                                                 scores, 0, gmax, den);
    }
    finalize<<<(NN * NF + 255) / 256, 256, 0, stream>>>(out, den);
}